// GAT_14946486190732
// MI455X (gfx1250) — compile-verified
//
#include <hip/hip_runtime.h>

// GAT on a chain graph: every destination has exactly one in-edge, so
// softmax alpha == 1.0 exactly and the whole attention path cancels.
// Remaining math:
//   out[b,0,:] = loc[b,0,:]
//   out[b,t,:] = loc[b,t-1,:] @ Wm^T + loc[b,t,:] @ Wr^T + bm   (t = 1..L-1)
// where Wm = mean_h(W_src), Wr = mean_h(W_res), bm = mean_h(bias).
// Bandwidth-bound (~134MB traffic, ~8.6 GFLOP fp32) -> V_WMMA_F32_16X16X4_F32.

typedef float v2f __attribute__((ext_vector_type(2)));
typedef float v8f __attribute__((ext_vector_type(8)));

#define FDIM 128
#define LDIM 4096
#define BDIM 32
#define HNUM 8

// ---- Kernel 1: head-mean of weights/bias into workspace ----
// ws layout (floats): [0, 16384)      Wm[o][e]
//                     [16384, 32768)  Wr[o][e]
//                     [32768, 32896)  bm[o]
__global__ __launch_bounds__(256) void gat_wred(const float* __restrict__ Wsrc,
                                                const float* __restrict__ Wres,
                                                const float* __restrict__ bias,
                                                float* __restrict__ ws) {
    const int idx = blockIdx.x * blockDim.x + threadIdx.x;
    const float inv = 1.0f / (float)HNUM;
    if (idx < FDIM * FDIM) {
        const int o = idx >> 7;
        const int e = idx & 127;
        float s0 = 0.0f, s1 = 0.0f;
#pragma unroll
        for (int h = 0; h < HNUM; ++h) {
            s0 += Wsrc[(h * FDIM + o) * FDIM + e];
            s1 += Wres[(h * FDIM + o) * FDIM + e];
        }
        ws[idx] = s0 * inv;
        ws[FDIM * FDIM + idx] = s1 * inv;
    }
    if (idx < FDIM) {
        float sb = 0.0f;
#pragma unroll
        for (int h = 0; h < HNUM; ++h) sb += bias[h * FDIM + idx];
        ws[2 * FDIM * FDIM + idx] = sb * inv;
    }
}

// ---- Kernel 2: out[b,0,:] = loc[b,0,:] ----
__global__ void gat_copy0(const float* __restrict__ loc, float* __restrict__ out) {
    const size_t base = (size_t)blockIdx.x * LDIM * FDIM;
    out[base + threadIdx.x] = loc[base + threadIdx.x];
}

// ---- Kernel 3: main WMMA GEMM ----
// One wave computes a 16(M) x 128(N) output strip for one batch.
// 256 row-tiles per batch cover output rows 1..4095 (last tile partially masked
// on store; loads for the phantom row are clamped in-range so EXEC stays all-1s
// through the WMMAs as the ISA requires).
__global__ __launch_bounds__(256) void gat_main(const float* __restrict__ loc,
                                                const float* __restrict__ ws,
                                                float* __restrict__ out) {
    const float* __restrict__ Wm = ws;
    const float* __restrict__ Wr = ws + FDIM * FDIM;
    const float* __restrict__ bm = ws + 2 * FDIM * FDIM;

    const int wave = threadIdx.x >> 5;
    const int lane = threadIdx.x & 31;
    const int gw   = blockIdx.x * (blockDim.x >> 5) + wave;
    const int b    = gw >> 8;                // 256 tiles per batch
    const int it   = gw & 255;
    const int row0 = 1 + (it << 4);          // first output row of this tile

    const int lrow  = lane & 15;             // M row / N col within tile
    const int khalf = lane >> 4;             // 0: K={k,k+1}, 1: K={k+2,k+3}

    const float* __restrict__ Xb = loc + (size_t)b * LDIM * FDIM;
    const int r  = row0 + lrow;
    const int rr = (r < LDIM - 1) ? r : (LDIM - 1);  // clamp phantom row loads
    const float* aS = Xb + (size_t)(rr - 1) * FDIM + 2 * khalf;  // src stream
    const float* aR = Xb + (size_t)rr       * FDIM + 2 * khalf;  // res stream

    v8f acc[8];
#pragma unroll
    for (int nt = 0; nt < 8; ++nt) acc[nt] = (v8f)(0.0f);

#pragma unroll 4
    for (int k = 0; k < FDIM; k += 4) {
        // A 16x4 fp32: lanes 0-15 hold K={k,k+1}, lanes 16-31 K={k+2,k+3}
        const v2f as = *(const v2f*)(aS + k);
        const v2f ar = *(const v2f*)(aR + k);
#pragma unroll
        for (int nt = 0; nt < 8; ++nt) {
            const int n = lrow + 16 * nt;
            // B 4x16 fp32 (symmetric layout): row-major Wm[n][k'] pairs
            const v2f bs = *(const v2f*)(Wm + n * FDIM + k + 2 * khalf);
            const v2f br = *(const v2f*)(Wr + n * FDIM + k + 2 * khalf);
            acc[nt] = __builtin_amdgcn_wmma_f32_16x16x4_f32(
                false, as, false, bs, (short)0, acc[nt], false, false);
            acc[nt] = __builtin_amdgcn_wmma_f32_16x16x4_f32(
                false, ar, false, br, (short)0, acc[nt], false, false);
        }
    }

    // C/D layout: VGPR j -> M = j + 8*khalf, N = lrow + 16*nt
    float* __restrict__ Ob = out + (size_t)b * LDIM * FDIM;
#pragma unroll
    for (int nt = 0; nt < 8; ++nt) {
        const int n = lrow + 16 * nt;
        const float bv = bm[n];
#pragma unroll
        for (int j = 0; j < 8; ++j) {
            const int orow = row0 + j + 8 * khalf;
            if (orow < LDIM) {
                Ob[(size_t)orow * FDIM + n] = acc[nt][j] + bv;
            }
        }
    }
}

extern "C" void kernel_launch(void* const* d_in, const int* in_sizes, int n_in,
                              void* d_out, int out_size, void* d_ws, size_t ws_size,
                              hipStream_t stream) {
    const float* loc  = (const float*)d_in[0];
    const float* Wsrc = (const float*)d_in[1];
    // d_in[2] W_dst, d_in[3] attn_l, d_in[4] attn_r: provably cancel (alpha==1).
    const float* Wres = (const float*)d_in[5];
    const float* bias = (const float*)d_in[6];
    float* ws  = (float*)d_ws;
    float* out = (float*)d_out;

    gat_wred<<<(FDIM * FDIM + 255) / 256, 256, 0, stream>>>(Wsrc, Wres, bias, ws);
    gat_copy0<<<BDIM, FDIM, 0, stream>>>(loc, out);
    // 32 batches * 256 tiles = 8192 waves; 8 waves (256 thr) per block.
    gat_main<<<(BDIM * 256) / 8, 256, 0, stream>>>(loc, ws, out);
}